// Long_encoder_2302102470699
// MI455X (gfx1250) — compile-verified
//
#include <hip/hip_runtime.h>
#include <math.h>

typedef float v2f __attribute__((ext_vector_type(2)));
typedef float v8f __attribute__((ext_vector_type(8)));

#define M_ROWS 14224   // 56 sequences * 254 patches
#define SEQS   56
#define TLEN   254
#define DM     128
#define DI     256
#define DFF    256
#define NDBL   40      // DT_RANK(8) + 2*D_STATE(32)

enum { EPI_NONE = 0, EPI_BIAS = 1, EPI_GELU = 2, EPI_SOFTPLUS = 3 };

__device__ __forceinline__ float sigmoid_(float x) { return 1.f / (1.f + expf(-x)); }

// ---------------------------------------------------------------------------
// Generic fp32 WMMA GEMM:  C[M,N] = A[M,K] @ W[N,K]^T  (+ bias, + activation)
// All shape parameters are compile-time: loads use immediate offsets from a
// single base pointer, no remainder loop, no pointer-increment bookkeeping.
// Each wave computes TWO adjacent 16x16 tiles (shared A fragment, two
// independent WMMA accumulator chains) with V_WMMA_F32_16X16X4_F32.
// ---------------------------------------------------------------------------
template <int EPI, bool NGUARD, int K, int LDA, int N, int LDC>
__global__ void gemm_wmma_f32(const float* __restrict__ A,
                              const float* __restrict__ W,       // (N,K) row-major
                              const float* __restrict__ bias,
                              float* __restrict__ C)
{
    const int wave  = threadIdx.x >> 5;
    const int lane  = threadIdx.x & 31;
    const int hi    = lane >> 4;      // half-wave select
    const int r     = lane & 15;
    const int mtile = blockIdx.y * 4 + wave;
    if (mtile * 16 >= M_ROWS) return; // wave-uniform: EXEC stays all-1s
    const int m0 = mtile * 16;
    const int n0 = blockIdx.x * 32;

    const int  nA  = n0 + r;
    const int  nB  = n0 + 16 + r;
    const bool okA = (!NGUARD) || (nA < N);
    const bool okB = (!NGUARD) || (nB < N);
    const int  rA  = NGUARD ? (okA ? nA : N - 1) : nA;   // clamp: load stays in-bounds
    const int  rB  = NGUARD ? (okB ? nB : N - 1) : nB;

    v8f acc0, acc1;
#pragma unroll
    for (int v = 0; v < 8; ++v) { acc0[v] = 0.f; acc1[v] = 0.f; }

    const float* arow  = A + (size_t)(m0 + r) * LDA + 2 * hi;
    const float* wrowA = W + (size_t)rA * K + 2 * hi;
    const float* wrowB = W + (size_t)rB * K + 2 * hi;

#pragma unroll (K >= 32 ? 8 : K / 4)
    for (int k0 = 0; k0 < K; k0 += 4) {
        v2f a  = *reinterpret_cast<const v2f*>(arow  + k0);
        v2f b0 = *reinterpret_cast<const v2f*>(wrowA + k0);
        v2f b1 = *reinterpret_cast<const v2f*>(wrowB + k0);
        if (NGUARD) {                 // branchless v_cndmask, not exec-masked loads
            b0[0] = okA ? b0[0] : 0.f;  b0[1] = okA ? b0[1] : 0.f;
            b1[0] = okB ? b1[0] : 0.f;  b1[1] = okB ? b1[1] : 0.f;
        }
        acc0 = __builtin_amdgcn_wmma_f32_16x16x4_f32(false, a, false, b0,
                                                     (short)0, acc0, false, false);
        acc1 = __builtin_amdgcn_wmma_f32_16x16x4_f32(false, a, false, b1,
                                                     (short)0, acc1, false, false);
    }

    const float bvA = (EPI != EPI_NONE && okA) ? bias[nA] : 0.f;
    const float bvB = (EPI != EPI_NONE && okB) ? bias[nB] : 0.f;
    float* crow = C + (size_t)(m0 + 8 * hi) * LDC;
#pragma unroll
    for (int v = 0; v < 8; ++v) {
        float t0 = acc0[v] + bvA;
        float t1 = acc1[v] + bvB;
        if (EPI == EPI_GELU) {
            t0 = 0.5f * t0 * (1.f + erff(t0 * 0.70710678118654752f));
            t1 = 0.5f * t1 * (1.f + erff(t1 * 0.70710678118654752f));
        } else if (EPI == EPI_SOFTPLUS) {
            t0 = fmaxf(t0, 0.f) + log1pf(expf(-fabsf(t0)));
            t1 = fmaxf(t1, 0.f) + log1pf(expf(-fabsf(t1)));
        }
        if (okA) crow[(size_t)v * LDC + nA] = t0;
        if (okB) crow[(size_t)v * LDC + nB] = t1;
    }
}

// ---------------------------------------------------------------------------
// Patch gather: X0[m][k] = z[seq][p*STRIDE + k],  m = seq*254 + p
// ---------------------------------------------------------------------------
__global__ void gather_patches(const float* __restrict__ z, float* __restrict__ X0, int total)
{
    int i = blockIdx.x * blockDim.x + threadIdx.x;
    if (i >= total) return;
    int k = i & 15, m = i >> 4;
    int seq = m / TLEN, p = m % TLEN;
    X0[i] = z[(size_t)seq * 2048 + p * 8 + k];
}

// ---------------------------------------------------------------------------
// Depthwise causal conv (width 4, left pad 3) + SiLU.  xm = XZ[:, 0:256]
// ---------------------------------------------------------------------------
__global__ void conv_silu(const float* __restrict__ XZ,
                          const float* __restrict__ cw,   // (256,4)
                          const float* __restrict__ cb,   // (256)
                          float* __restrict__ U)
{
    int m = blockIdx.x, d = threadIdx.x;
    int t = m % TLEN;
    const float* base = XZ + (size_t)m * 512 + d;
    float acc = cb[d] + cw[d * 4 + 3] * base[0];
    if (t >= 1) acc += cw[d * 4 + 2] * base[-512];
    if (t >= 2) acc += cw[d * 4 + 1] * base[-1024];
    if (t >= 3) acc += cw[d * 4 + 0] * base[-1536];
    U[(size_t)m * DI + d] = acc * sigmoid_(acc);
}

// ---------------------------------------------------------------------------
// Selective scan + skip (u*D) + SiLU gate.  One block per sequence, one lane
// per channel; 16-state h lives in VGPRs.  Overwrites DELTA in place with yg.
// ---------------------------------------------------------------------------
__global__ void scan_gate(const float* __restrict__ U,
                          float* __restrict__ DELTA,          // in: delta, out: gated y
                          const float* __restrict__ DBLp,     // (M,40): [dt|B|C]
                          const float* __restrict__ XZ,       // zg = XZ[:,256:512]
                          const float* __restrict__ Alog,     // (256,16)
                          const float* __restrict__ Dp_)
{
    int seq = blockIdx.x, d = threadIdx.x;
    float An[16], h[16];
#pragma unroll
    for (int s = 0; s < 16; ++s) { An[s] = -expf(Alog[d * 16 + s]); h[s] = 0.f; }
    const float Dp = Dp_[d];

    size_t m = (size_t)seq * TLEN;
    for (int t = 0; t < TLEN; ++t, ++m) {
        float dl = DELTA[m * DI + d];
        float uu = U[m * DI + d];
        float Bl[16], Cl[16];
        const float4* b4 = reinterpret_cast<const float4*>(DBLp + m * NDBL + 8);
        const float4* c4 = reinterpret_cast<const float4*>(DBLp + m * NDBL + 24);
#pragma unroll
        for (int q = 0; q < 4; ++q) {
            reinterpret_cast<float4*>(Bl)[q] = b4[q];
            reinterpret_cast<float4*>(Cl)[q] = c4[q];
        }
        float y = 0.f, dlu = dl * uu;
#pragma unroll
        for (int s = 0; s < 16; ++s) {
            float dA = expf(dl * An[s]);
            h[s] = dA * h[s] + dlu * Bl[s];
            y += h[s] * Cl[s];
        }
        y += uu * Dp;
        float zg = XZ[m * 512 + 256 + d];
        DELTA[m * DI + d] = y * (zg * sigmoid_(zg));
    }
}

// ---------------------------------------------------------------------------
// out[b,c,dm,p] = X[(b*7+c)*254 + p][dm]
// ---------------------------------------------------------------------------
__global__ void write_out(const float* __restrict__ X, float* __restrict__ out)
{
    int m = blockIdx.x, dm = threadIdx.x;
    int seq = m / TLEN, p = m % TLEN;
    out[((size_t)seq * DM + dm) * TLEN + p] = X[(size_t)m * DM + dm];
}

// ---------------------------------------------------------------------------
extern "C" void kernel_launch(void* const* d_in, const int* in_sizes, int n_in,
                              void* d_out, int out_size, void* d_ws, size_t ws_size,
                              hipStream_t stream)
{
    const float* z    = (const float*)d_in[0];
    const float* WP   = (const float*)d_in[1];
    const float* WPb  = (const float*)d_in[2];
    const float* inW  = (const float*)d_in[3];
    const float* cw   = (const float*)d_in[4];
    const float* cb   = (const float*)d_in[5];
    const float* xw   = (const float*)d_in[6];
    const float* dtw  = (const float*)d_in[7];
    const float* dtb  = (const float*)d_in[8];
    const float* Alog = (const float*)d_in[9];
    const float* Dp   = (const float*)d_in[10];
    const float* ow   = (const float*)d_in[11];
    const float* l1w  = (const float*)d_in[12];
    const float* l1b  = (const float*)d_in[13];
    const float* l2w  = (const float*)d_in[14];
    const float* l2b  = (const float*)d_in[15];
    float* out = (float*)d_out;

    float* ws = (float*)d_ws;
    size_t o = 0;
    float* X   = ws + o; o += (size_t)M_ROWS * DM;    // 128-wide activations
    float* XZ  = ws + o; o += (size_t)M_ROWS * 512;   // in_proj out; later FFN hidden
    float* U   = ws + o; o += (size_t)M_ROWS * DI;    // conv+silu out
    float* DEL = ws + o; o += (size_t)M_ROWS * DI;    // delta, then gated scan output
    float* DBL = ws + o; o += (size_t)M_ROWS * NDBL;  // [dt|B|C]
    float* X2  = ws + o; o += (size_t)M_ROWS * DM;    // mamba block output
    float* X0  = ws + o; o += (size_t)M_ROWS * 16;    // gathered patches

    const dim3 blk(128, 1, 1);
    const int  mblocks = (M_ROWS / 16 + 3) / 4;       // 223 (last block: 1 live wave)

    gather_patches<<<(M_ROWS * 16 + 255) / 256, 256, 0, stream>>>(z, X0, M_ROWS * 16);
    // patch embed: K=16, N=128
    gemm_wmma_f32<EPI_BIAS, false, 16, 16, DM, DM>
        <<<dim3(DM / 32, mblocks), blk, 0, stream>>>(X0, WP, WPb, X);

    for (int i = 0; i < 3; ++i) {
        const float* inWi = inW  + (size_t)i * 512 * 128;
        const float* cwi  = cw   + (size_t)i * DI * 4;
        const float* cbi  = cb   + (size_t)i * DI;
        const float* xwi  = xw   + (size_t)i * NDBL * DI;
        const float* dtwi = dtw  + (size_t)i * DI * 8;
        const float* dtbi = dtb  + (size_t)i * DI;
        const float* Ali  = Alog + (size_t)i * DI * 16;
        const float* Dpi  = Dp   + (size_t)i * DI;
        const float* owi  = ow   + (size_t)i * DM * DI;
        const float* l1wi = l1w  + (size_t)i * DFF * DM;
        const float* l1bi = l1b  + (size_t)i * DFF;
        const float* l2wi = l2w  + (size_t)i * DM * DFF;
        const float* l2bi = l2b  + (size_t)i * DM;

        // in_proj: K=128, N=512
        gemm_wmma_f32<EPI_NONE, false, 128, 128, 512, 512>
            <<<dim3(512 / 32, mblocks), blk, 0, stream>>>(X, inWi, nullptr, XZ);
        conv_silu<<<M_ROWS, DI, 0, stream>>>(XZ, cwi, cbi, U);
        // x_proj: K=256, N=40 (guarded)
        gemm_wmma_f32<EPI_NONE, true, 256, 256, NDBL, NDBL>
            <<<dim3(2, mblocks), blk, 0, stream>>>(U, xwi, nullptr, DBL);
        // dt_proj + softplus: K=8 (lda=40), N=256
        gemm_wmma_f32<EPI_SOFTPLUS, false, 8, NDBL, DI, DI>
            <<<dim3(DI / 32, mblocks), blk, 0, stream>>>(DBL, dtwi, dtbi, DEL);
        scan_gate<<<SEQS, DI, 0, stream>>>(U, DEL, DBL, XZ, Ali, Dpi);
        // out_proj: K=256, N=128
        gemm_wmma_f32<EPI_NONE, false, 256, 256, DM, DM>
            <<<dim3(DM / 32, mblocks), blk, 0, stream>>>(DEL, owi, nullptr, X2);
        // lin1 + gelu: K=128, N=256  (hidden H reuses XZ buffer)
        gemm_wmma_f32<EPI_GELU, false, 128, 128, DFF, DFF>
            <<<dim3(DFF / 32, mblocks), blk, 0, stream>>>(X2, l1wi, l1bi, XZ);
        // lin2 + bias: K=256, N=128
        gemm_wmma_f32<EPI_BIAS, false, 256, 256, DM, DM>
            <<<dim3(DM / 32, mblocks), blk, 0, stream>>>(XZ, l2wi, l2bi, X);
    }

    write_out<<<M_ROWS, DM, 0, stream>>>(X, out);
}